// EncoderWithoutFlexFrontsGAT_26139170964035
// MI455X (gfx1250) — compile-verified
//
#include <hip/hip_runtime.h>
#include <math.h>

// ---------------------------------------------------------------------------
// GAT x3 for MI455X (gfx1250, wave32).
// GEMMs run on V_WMMA_F32_16X16X4_F32 (exact fp32, matrix path is not the
// bottleneck); edge phase is L2-resident atomics (segment max/sum/scatter).
// ---------------------------------------------------------------------------

typedef __attribute__((ext_vector_type(2))) float v2f;
typedef __attribute__((ext_vector_type(8))) float v8f;

#define NEG_SLOPE 0.2f

// ---------------- GEMM: C[M,64] = A[M,K] @ W[K,64] via fp32 WMMA ------------
// One wave -> one 16x16 tile. 8 waves per block -> 32 rows x 64 cols.
__global__ __launch_bounds__(256) void gat_gemm_wmma_f32(
    const float* __restrict__ A, const float* __restrict__ W,
    float* __restrict__ C, int M, int K) {
  const int wave = threadIdx.x >> 5;         // 0..7
  const int lane = threadIdx.x & 31;
  const int ntile = wave & 3;                // column tile 0..3 (64 cols)
  const int mtile = (blockIdx.x << 1) | (wave >> 2);
  if (mtile * 16 >= M) return;

  const int mrow = mtile * 16 + (lane & 15); // A row this lane feeds
  const int koff = (lane >> 4) << 1;         // lanes 0-15: K=0,1; 16-31: K=2,3
  const int col  = ntile * 16 + (lane & 15); // B/D column this lane feeds

  const float* __restrict__ arow = A + (size_t)mrow * K;
  v8f acc = {};
  for (int k = 0; k < K; k += 4) {
    v2f a, b;
    a.x = arow[k + koff];
    a.y = arow[k + koff + 1];
    b.x = W[(size_t)(k + koff)     * 64 + col];
    b.y = W[(size_t)(k + koff + 1) * 64 + col];
    // (neg_a, A, neg_b, B, c_mod, C, reuse_a, reuse_b)
    acc = __builtin_amdgcn_wmma_f32_16x16x4_f32(
        false, a, false, b, (short)0, acc, false, false);
  }
  // D layout: lanes 0-15 -> rows v (v=0..7); lanes 16-31 -> rows v+8.
  const int rbase = mtile * 16 + ((lane >> 4) << 3);
#pragma unroll
  for (int v = 0; v < 8; ++v)
    C[(size_t)(rbase + v) * 64 + col] = acc[v];
}

// -------- per-node alpha_s / alpha_d: one wave per node, 64 channels --------
__global__ __launch_bounds__(256) void gat_alpha(
    const float* __restrict__ hp, const float* __restrict__ a_s,
    const float* __restrict__ a_d, float* __restrict__ out_s,
    float* __restrict__ out_d, int N) {
  const int node = blockIdx.x * 8 + (threadIdx.x >> 5);
  const int lane = threadIdx.x & 31;
  if (node >= N) return;
  const float* __restrict__ h = hp + (size_t)node * 64;
  float s = h[lane] * a_s[lane] + h[lane + 32] * a_s[lane + 32];
  float d = h[lane] * a_d[lane] + h[lane + 32] * a_d[lane + 32];
#pragma unroll
  for (int off = 16; off > 0; off >>= 1) {
    s += __shfl_down(s, off, 32);
    d += __shfl_down(d, off, 32);
  }
  if (lane == 0) { out_s[node] = s; out_d[node] = d; }
}

// ------------------------------ init per layer ------------------------------
__global__ void gat_init(float* __restrict__ accum, unsigned int* __restrict__ maxk,
                         float* __restrict__ denom, int N) {
  const int tid = blockIdx.x * blockDim.x + threadIdx.x;
  if (tid < N * 64) accum[tid] = 0.0f;
  if (tid < N) { maxk[tid] = 0u; denom[tid] = 0.0f; }
}

// order-preserving float<->uint for atomic max (init key 0 < all encodings)
__device__ __forceinline__ unsigned int f2key(float f) {
  unsigned int u = __float_as_uint(f);
  return (u & 0x80000000u) ? ~u : (u | 0x80000000u);
}
__device__ __forceinline__ float key2f(unsigned int k) {
  return __uint_as_float((k & 0x80000000u) ? (k ^ 0x80000000u) : ~k);
}

__device__ __forceinline__ void edge_ends(const int* __restrict__ ei, int idx,
                                          int E, int& src, int& dst) {
  if (idx < E) { src = ei[idx]; dst = ei[E + idx]; }
  else         { src = idx - E; dst = idx - E; }   // appended self-loops
}

__device__ __forceinline__ float edge_logit(const float* __restrict__ as_,
                                            const float* __restrict__ ad_,
                                            int src, int dst) {
  float e = as_[src] + ad_[dst];
  return (e > 0.0f) ? e : NEG_SLOPE * e;
}

// ----------------------- pass 1: segment max over dst -----------------------
__global__ void gat_edge_max(const int* __restrict__ ei,
                             const float* __restrict__ as_,
                             const float* __restrict__ ad_,
                             unsigned int* __restrict__ maxk, int E, int N) {
  const int idx = blockIdx.x * blockDim.x + threadIdx.x;
  if (idx >= E + N) return;
  int src, dst; edge_ends(ei, idx, E, src, dst);
  atomicMax(&maxk[dst], f2key(edge_logit(as_, ad_, src, dst)));
}

// ------------------- pass 2: exp(e - max) and segment sum -------------------
__global__ void gat_edge_exp(const int* __restrict__ ei,
                             const float* __restrict__ as_,
                             const float* __restrict__ ad_,
                             const unsigned int* __restrict__ maxk,
                             float* __restrict__ denom,
                             float* __restrict__ eexp, int E, int N) {
  const int idx = blockIdx.x * blockDim.x + threadIdx.x;
  if (idx >= E + N) return;
  int src, dst; edge_ends(ei, idx, E, src, dst);
  const float v = __expf(edge_logit(as_, ad_, src, dst) - key2f(maxk[dst]));
  eexp[idx] = v;
  atomicAdd(&denom[dst], v);
}

// ------------- pass 3: attention-weighted scatter of messages ---------------
__global__ void gat_edge_scatter(const int* __restrict__ ei,
                                 const float* __restrict__ hp,
                                 const float* __restrict__ eexp,
                                 const float* __restrict__ denom,
                                 float* __restrict__ accum, int E, int N) {
  const long long tid = (long long)blockIdx.x * blockDim.x + threadIdx.x;
  if (tid >= (long long)(E + N) * 64) return;
  const int idx = (int)(tid >> 6);
  const int c   = (int)(tid & 63);
  int src, dst; edge_ends(ei, idx, E, src, dst);
  const float a = eexp[idx] / (denom[dst] + 1e-16f);
  atomicAdd(&accum[(size_t)dst * 64 + c], hp[(size_t)src * 64 + c] * a);
}

// ------------------------- bias + ReLU finalize -----------------------------
__global__ void gat_finalize(const float* __restrict__ accum,
                             const float* __restrict__ b,
                             float* __restrict__ out, int N) {
  const int tid = blockIdx.x * blockDim.x + threadIdx.x;
  if (tid >= N * 64) return;
  const float v = accum[tid] + b[tid & 63];
  out[tid] = v > 0.0f ? v : 0.0f;
}

// ---------------------------------------------------------------------------
extern "C" void kernel_launch(void* const* d_in, const int* in_sizes, int n_in,
                              void* d_out, int out_size, void* d_ws, size_t ws_size,
                              hipStream_t stream) {
  (void)n_in; (void)out_size; (void)ws_size;

  const float* x   = (const float*)d_in[0];
  const int*   ei  = (const int*)d_in[1];
  const int N = in_sizes[0] / 128;      // 50000
  const int E = in_sizes[1] / 2;        // 800000
  const int T = E + N;                  // edges incl. self-loops

  // workspace layout (floats)
  float* ws    = (float*)d_ws;
  float* hp    = ws;                         // N*64
  float* hcur  = hp    + (size_t)N * 64;     // N*64
  float* accum = hcur  + (size_t)N * 64;     // N*64
  float* as_   = accum + (size_t)N * 64;     // N
  float* ad_   = as_   + N;                  // N
  unsigned int* maxk = (unsigned int*)(ad_ + N); // N
  float* denom = (float*)(maxk + N);         // N
  float* eexp  = denom + N;                  // E+N

  const float* hin = x;
  int K = 128;
  for (int layer = 0; layer < 3; ++layer) {
    const float* W    = (const float*)d_in[4 + layer * 4];
    const float* asrc = (const float*)d_in[5 + layer * 4];
    const float* adst = (const float*)d_in[6 + layer * 4];
    const float* bias = (const float*)d_in[7 + layer * 4];

    // hp = hin @ W  (WMMA fp32)
    const int mblocks = (N / 16 + 1) / 2;    // 2 row-tiles per block
    gat_gemm_wmma_f32<<<mblocks + 1, 256, 0, stream>>>(hin, W, hp, N, K);

    // per-node attention coefficients
    gat_alpha<<<(N + 7) / 8, 256, 0, stream>>>(hp, asrc, adst, as_, ad_, N);

    // segment softmax + scatter
    gat_init<<<(N * 64 + 255) / 256, 256, 0, stream>>>(accum, maxk, denom, N);
    gat_edge_max<<<(T + 255) / 256, 256, 0, stream>>>(ei, as_, ad_, maxk, E, N);
    gat_edge_exp<<<(T + 255) / 256, 256, 0, stream>>>(ei, as_, ad_, maxk, denom, eexp, E, N);
    {
      long long tot = (long long)T * 64;
      gat_edge_scatter<<<(unsigned int)((tot + 255) / 256), 256, 0, stream>>>(
          ei, hp, eexp, denom, accum, E, N);
    }

    float* dst = (layer == 2) ? (float*)d_out : hcur;
    gat_finalize<<<(N * 64 + 255) / 256, 256, 0, stream>>>(accum, bias, dst, N);

    hin = hcur;
    K = 64;
  }
}